// GNN_Layer_27058293965313
// MI455X (gfx1250) — compile-verified
//
#include <hip/hip_runtime.h>

// Problem constants (match reference)
#define NN 50000
#define FF 128
#define CC 128
#define KK 10

typedef __attribute__((ext_vector_type(2))) float v2f;
typedef __attribute__((ext_vector_type(8))) float v8f;

// ---------------------------------------------------------------------------
// GEMM kernel: computes three N x C products of Z[N][F] with Wsv/Wsr/Wdr
// using V_WMMA_F32_16X16X4_F32.  Block = 256 threads (8 wave32).
// Block handles 16 output rows; wave w handles output cols [16w, 16w+16).
// node_signals -> outV (pre-activation, gather kernel finishes it)
// Z@Wsr -> tS,  Z@Wdr -> tD  (workspace tables, gathered later)
// ---------------------------------------------------------------------------
__global__ __launch_bounds__(256) void gnn_gemm3(
    const float* __restrict__ Z,
    const float* __restrict__ Wsv,
    const float* __restrict__ Wsr,
    const float* __restrict__ Wdr,
    float* __restrict__ outV,
    float* __restrict__ tS,
    float* __restrict__ tD)
{
    // 16 x 128 tile of Z, padded stride 132 for conflict-free b64 reads
    __shared__ float Zs[16][132];

    const int t    = threadIdx.x;
    const int wave = t >> 5;
    const int lane = t & 31;
    const int half = lane >> 4;   // 0: lanes 0-15, 1: lanes 16-31
    const int lr   = lane & 15;
    const int m0   = blockIdx.x * 16;   // N % 16 == 0, no tail
    const int n0   = wave * 16;         // 8 waves cover C = 128

    // Cooperative LDS fill: 16*128 floats = 512 float4, 2 per thread.
    {
        const float4* __restrict__ src =
            reinterpret_cast<const float4*>(Z + (size_t)m0 * FF);
#pragma unroll
        for (int i = 0; i < 2; ++i) {
            int idx = t + i * 256;          // 0..511
            float4 v = src[idx];
            int row = idx >> 5;             // 32 float4 per row
            int col = (idx & 31) << 2;
            Zs[row][col + 0] = v.x;
            Zs[row][col + 1] = v.y;
            Zs[row][col + 2] = v.z;
            Zs[row][col + 3] = v.w;
        }
    }
    __syncthreads();

    v8f acc_v = {};
    v8f acc_s = {};
    v8f acc_d = {};

    // K loop: F = 128 in steps of 4.
#pragma unroll 8
    for (int kk = 0; kk < FF / 4; ++kk) {
        const int k0 = kk * 4;
        // A fragment (16x4 f32): lanes 0-15 hold K = k0,k0+1; lanes 16-31 hold K = k0+2,k0+3
        const v2f a = *reinterpret_cast<const v2f*>(&Zs[lr][k0 + 2 * half]);

        // B fragments (4x16 f32), mirrored layout:
        // VGPR0: row k0 (+2 for upper lane half), VGPR1: row k0+1 (+2)
        const int wrow = k0 + 2 * half;
        const int bcol = n0 + lr;
        v2f bv, bs, bd;
        bv.x = Wsv[(size_t)wrow * CC + bcol];
        bv.y = Wsv[(size_t)(wrow + 1) * CC + bcol];
        bs.x = Wsr[(size_t)wrow * CC + bcol];
        bs.y = Wsr[(size_t)(wrow + 1) * CC + bcol];
        bd.x = Wdr[(size_t)wrow * CC + bcol];
        bd.y = Wdr[(size_t)(wrow + 1) * CC + bcol];

        // D = A*B + C, fp32 exact vs reference
        acc_v = __builtin_amdgcn_wmma_f32_16x16x4_f32(
            false, a, false, bv, (short)0, acc_v, false, false);
        acc_s = __builtin_amdgcn_wmma_f32_16x16x4_f32(
            false, a, false, bs, (short)0, acc_s, false, false);
        acc_d = __builtin_amdgcn_wmma_f32_16x16x4_f32(
            false, a, false, bd, (short)0, acc_d, false, false);
    }

    // C/D layout: VGPR v holds row M = v + 8*half, col N = lr.
#pragma unroll
    for (int v = 0; v < 8; ++v) {
        const size_t row = (size_t)(m0 + v + 8 * half);
        const size_t off = row * CC + (n0 + lr);
        outV[off] = acc_v[v];
        tS[off]   = acc_s[v];
        tD[off]   = acc_d[v];
    }
}

// ---------------------------------------------------------------------------
// Gather kernel: 32 lanes per node, 8 nodes per 256-thread block.
// Lane handles channels [4*lane, 4*lane+4) as float4 (fully coalesced:
// one neighbor row = 512B per wave). Tables are L2-resident (25.6 MB each).
// out[i] = relu(out[i] + mean_valid(tS[same[i]]) + mean_valid(tD[diff[i]]))
// ---------------------------------------------------------------------------
__global__ __launch_bounds__(256) void gnn_gather(
    const int* __restrict__ same,
    const int* __restrict__ diff,
    const float* __restrict__ tS,
    const float* __restrict__ tD,
    float* __restrict__ out)
{
    const int lane = threadIdx.x & 31;
    const int grp  = threadIdx.x >> 5;
    const int node = blockIdx.x * 8 + grp;   // NN % 8 == 0
    if (node >= NN) return;

    float4 accS = make_float4(0.f, 0.f, 0.f, 0.f);
    float4 accD = make_float4(0.f, 0.f, 0.f, 0.f);
    int cs = 0, cd = 0;

#pragma unroll
    for (int j = 0; j < KK; ++j) {
        const int is = same[(size_t)node * KK + j];
        if (is > -1) {
            const float4 v = *reinterpret_cast<const float4*>(
                tS + (size_t)is * CC + 4 * lane);
            accS.x += v.x; accS.y += v.y; accS.z += v.z; accS.w += v.w;
            ++cs;
        }
        const int id = diff[(size_t)node * KK + j];
        if (id > -1) {
            const float4 v = *reinterpret_cast<const float4*>(
                tD + (size_t)id * CC + 4 * lane);
            accD.x += v.x; accD.y += v.y; accD.z += v.z; accD.w += v.w;
            ++cd;
        }
    }

    const float ns = 1.0f / (float)max(cs, 1);
    const float nd = 1.0f / (float)max(cd, 1);

    float* op = out + (size_t)node * CC + 4 * lane;
    float4 o = *reinterpret_cast<const float4*>(op);
    o.x = fmaxf(o.x + accS.x * ns + accD.x * nd, 0.0f);
    o.y = fmaxf(o.y + accS.y * ns + accD.y * nd, 0.0f);
    o.z = fmaxf(o.z + accS.z * ns + accD.z * nd, 0.0f);
    o.w = fmaxf(o.w + accS.w * ns + accD.w * nd, 0.0f);
    *reinterpret_cast<float4*>(op) = o;
}

// ---------------------------------------------------------------------------
// Launch: GEMM1 -> Gather1 -> GEMM2 -> Gather2 (stream-ordered, so both
// proteins reuse the same 51.2 MB of workspace tables).
// ---------------------------------------------------------------------------
extern "C" void kernel_launch(void* const* d_in, const int* in_sizes, int n_in,
                              void* d_out, int out_size, void* d_ws, size_t ws_size,
                              hipStream_t stream) {
    const float* Z1    = (const float*)d_in[0];
    const int*   same1 = (const int*)  d_in[1];
    const int*   diff1 = (const int*)  d_in[2];
    const float* Z2    = (const float*)d_in[3];
    const int*   same2 = (const int*)  d_in[4];
    const int*   diff2 = (const int*)  d_in[5];
    const float* Wsv   = (const float*)d_in[6];
    const float* Wsr   = (const float*)d_in[7];
    const float* Wdr   = (const float*)d_in[8];

    float* out1 = (float*)d_out;
    float* out2 = out1 + (size_t)NN * CC;

    float* tS = (float*)d_ws;
    float* tD = tS + (size_t)NN * CC;

    const dim3 gemmGrid(NN / 16), gemmBlock(256);
    const dim3 gatGrid(NN / 8),  gatBlock(256);

    gnn_gemm3<<<gemmGrid, gemmBlock, 0, stream>>>(Z1, Wsv, Wsr, Wdr, out1, tS, tD);
    gnn_gather<<<gatGrid, gatBlock, 0, stream>>>(same1, diff1, tS, tD, out1);

    gnn_gemm3<<<gemmGrid, gemmBlock, 0, stream>>>(Z2, Wsv, Wsr, Wdr, out2, tS, tD);
    gnn_gather<<<gatGrid, gatBlock, 0, stream>>>(same2, diff2, tS, tD, out2);
}